// DGCNLayer_32143535243482
// MI455X (gfx1250) — compile-verified
//
#include <hip/hip_runtime.h>
#include <hip/hip_bf16.h>

typedef float v2f __attribute__((ext_vector_type(2)));
typedef float v8f __attribute__((ext_vector_type(8)));

#define N_NODES 100000
#define N_EDGES 800000
#define D 128
#define ALPHA 0.1f

// ---------------------------------------------------------------------------
// Generic f32 WMMA GEMM:  C[M x 128] = act( [A1 | A2][M x KTOT] @ W[KTOT x 128] + bias )
// A1/A2 are row-major with 128 columns each; A2 supplies K in [128,256).
// One wave computes a 16-row x 128-col tile using V_WMMA_F32_16X16X4_F32.
// ---------------------------------------------------------------------------
template <int KTOT, bool ACT>
__global__ __launch_bounds__(128) void gemm_wmma_f32(
    const float* __restrict__ A1, const float* __restrict__ A2,
    const float* __restrict__ W, const float* __restrict__ bias,
    float* __restrict__ C, int M) {
  const int lane = threadIdx.x & 31;
  const int wave = threadIdx.x >> 5;
  const int rowTile = blockIdx.x * (blockDim.x >> 5) + wave;
  if (rowTile >= (M >> 4)) return;  // uniform per-wave exit; EXEC stays all-1s

  const int rrow = (lane & 15);                // row within tile for A frag
  const int koff = (lane >> 4) << 1;           // 0 (lanes 0-15) or 2 (lanes 16-31)
  const int ncol = lane & 15;                  // col within N-tile for B/C frags

  v8f acc[8] = {};

  const size_t aRow = (size_t)(rowTile * 16 + rrow) * 128;

  for (int k0 = 0; k0 < KTOT; k0 += 4) {
    const float* __restrict__ src = (KTOT == 256 && k0 >= 128) ? A2 : A1;
    const int kc = k0 & 127;
    // A fragment: 2 consecutive K values for this lane's row
    v2f a = *(const v2f*)(src + aRow + kc + koff);
    // B fragments per N-tile: W[(k0+koff)+v][ntile*16+ncol], v in {0,1}
    const float* __restrict__ wk = W + (size_t)(k0 + koff) * 128;
#pragma unroll
    for (int t = 0; t < 8; ++t) {
      v2f b;
      b.x = wk[t * 16 + ncol];
      b.y = wk[128 + t * 16 + ncol];
      acc[t] = __builtin_amdgcn_wmma_f32_16x16x4_f32(
          /*neg_a=*/false, a, /*neg_b=*/false, b,
          /*c_mod=*/(short)0, acc[t], /*reuse_a=*/false, /*reuse_b=*/false);
    }
  }

  // C/D layout: VGPR i -> row i (lanes 0-15) or i+8 (lanes 16-31), col = lane&15
  const int outRowBase = rowTile * 16 + ((lane >> 4) << 3);
#pragma unroll
  for (int t = 0; t < 8; ++t) {
    const int c = t * 16 + ncol;
    const float bb = ACT ? bias[c] : 0.f;
#pragma unroll
    for (int i = 0; i < 8; ++i) {
      float v = acc[t][i];
      if (ACT) v = fmaxf(v + bb, 0.f);  // no dead +0.0 in the non-ACT path
      C[(size_t)(outRowBase + i) * 128 + c] = v;
    }
  }
}

// ---------------------------------------------------------------------------
// Zero two aggregate buffers (float4-vectorized)
// ---------------------------------------------------------------------------
__global__ __launch_bounds__(256) void zero2_kernel(float* __restrict__ a,
                                                    float* __restrict__ b,
                                                    int n4) {
  int t = blockIdx.x * blockDim.x + threadIdx.x;
  if (t >= n4) return;
  float4 z = make_float4(0.f, 0.f, 0.f, 0.f);
  ((float4*)a)[t] = z;
  ((float4*)b)[t] = z;
}

// ---------------------------------------------------------------------------
// Edge scatter: 32 lanes per edge, 4 contiguous feature columns per lane.
//   aggU[rows[e]] += supV[cols[e]] * vals[e]
//   aggI[cols[e]] += supU[rows[e]] * vals[e]
// ---------------------------------------------------------------------------
__global__ __launch_bounds__(256) void scatter_edges_kernel(
    const int* __restrict__ rows, const int* __restrict__ cols,
    const float* __restrict__ vals, const float* __restrict__ supV,
    const float* __restrict__ supU, float* __restrict__ aggU,
    float* __restrict__ aggI, int E) {
  long long tid = (long long)blockIdx.x * blockDim.x + threadIdx.x;
  int e = (int)(tid >> 5);
  if (e >= E) return;
  const int c4 = ((int)tid & 31) << 2;
  const int u = rows[e];
  const int it = cols[e];
  const float v = vals[e];

  const float4 sv = *(const float4*)(supV + (size_t)it * 128 + c4);
  const float4 su = *(const float4*)(supU + (size_t)u * 128 + c4);
  float* du = aggU + (size_t)u * 128 + c4;
  float* di = aggI + (size_t)it * 128 + c4;
  atomicAdd(du + 0, sv.x * v);
  atomicAdd(du + 1, sv.y * v);
  atomicAdd(du + 2, sv.z * v);
  atomicAdd(du + 3, sv.w * v);
  atomicAdd(di + 0, su.x * v);
  atomicAdd(di + 1, su.y * v);
  atomicAdd(di + 2, su.z * v);
  atomicAdd(di + 3, su.w * v);
}

// ---------------------------------------------------------------------------
// In-place bias + LeakyReLU on two [N x 128] arrays (float4-vectorized)
// ---------------------------------------------------------------------------
__global__ __launch_bounds__(256) void bias_leaky_kernel(
    float* __restrict__ x1, const float* __restrict__ b1,
    float* __restrict__ x2, const float* __restrict__ b2, int n4) {
  int t = blockIdx.x * blockDim.x + threadIdx.x;
  if (t >= n4) return;
  size_t i = (size_t)t * 4;
  int col = (int)(i & 127);
  float4 bb;
  float4 v;

  bb = *(const float4*)(b1 + col);
  v = *(float4*)(x1 + i);
  v.x += bb.x; v.y += bb.y; v.z += bb.z; v.w += bb.w;
  v.x = (v.x >= 0.f) ? v.x : ALPHA * v.x;
  v.y = (v.y >= 0.f) ? v.y : ALPHA * v.y;
  v.z = (v.z >= 0.f) ? v.z : ALPHA * v.z;
  v.w = (v.w >= 0.f) ? v.w : ALPHA * v.w;
  *(float4*)(x1 + i) = v;

  bb = *(const float4*)(b2 + col);
  v = *(float4*)(x2 + i);
  v.x += bb.x; v.y += bb.y; v.z += bb.z; v.w += bb.w;
  v.x = (v.x >= 0.f) ? v.x : ALPHA * v.x;
  v.y = (v.y >= 0.f) ? v.y : ALPHA * v.y;
  v.z = (v.z >= 0.f) ? v.z : ALPHA * v.z;
  v.w = (v.w >= 0.f) ? v.w : ALPHA * v.w;
  *(float4*)(x2 + i) = v;
}

// ---------------------------------------------------------------------------
// Passthrough copies of ufea/vfea into output slices 2 and 5
// ---------------------------------------------------------------------------
__global__ __launch_bounds__(256) void copy2_kernel(const float* __restrict__ s1,
                                                    float* __restrict__ d1,
                                                    const float* __restrict__ s2,
                                                    float* __restrict__ d2,
                                                    int n4) {
  int t = blockIdx.x * blockDim.x + threadIdx.x;
  if (t >= n4) return;
  ((float4*)d1)[t] = ((const float4*)s1)[t];
  ((float4*)d2)[t] = ((const float4*)s2)[t];
}

extern "C" void kernel_launch(void* const* d_in, const int* in_sizes, int n_in,
                              void* d_out, int out_size, void* d_ws,
                              size_t ws_size, hipStream_t stream) {
  const float* ufea = (const float*)d_in[0];
  const float* vfea = (const float*)d_in[1];
  const int* uv_rows = (const int*)d_in[2];
  const int* uv_cols = (const int*)d_in[3];
  const float* uv_vals = (const float*)d_in[4];
  const float* W1 = (const float*)d_in[5];
  const float* b1 = (const float*)d_in[6];
  const float* W2 = (const float*)d_in[7];
  const float* b2 = (const float*)d_in[8];
  const float* Wu = (const float*)d_in[9];
  const float* bu = (const float*)d_in[10];
  const float* Wi = (const float*)d_in[11];
  const float* bi = (const float*)d_in[12];

  float* out = (float*)d_out;
  const size_t S = (size_t)N_NODES * D;  // 12.8M floats per slice
  float* s0 = out + 0 * S;  // relu(User_h)   (temp: support_v = vfea@W1)
  float* s1 = out + 1 * S;  // User_n         (agg + bias + leaky, in place)
  float* s2 = out + 2 * S;  // ufea copy
  float* s3 = out + 3 * S;  // relu(Item_h)   (temp: support_u = ufea@W2)
  float* s4 = out + 4 * S;  // Item_n
  float* s5 = out + 5 * S;  // vfea copy

  const int n4 = (int)(S / 4);                        // 3.2M float4 per slice
  const int elemBlocks = (n4 + 255) / 256;            // 12500
  const int gemmBlocks = ((N_NODES / 16) + 3) / 4;    // 6250 tiles / 4 waves
  const long long scatterThreads = (long long)N_EDGES * 32;
  const int scatterBlocks = (int)((scatterThreads + 255) / 256);  // 100000

  // 1) zero the aggregation targets (output slices 1 and 4)
  zero2_kernel<<<elemBlocks, 256, 0, stream>>>(s1, s4, n4);

  // 2) support matrices via f32 WMMA (no bias / no activation)
  gemm_wmma_f32<128, false><<<gemmBlocks, 128, 0, stream>>>(
      vfea, nullptr, W1, nullptr, s0, N_NODES);
  gemm_wmma_f32<128, false><<<gemmBlocks, 128, 0, stream>>>(
      ufea, nullptr, W2, nullptr, s3, N_NODES);

  // 3) edge gather/scale/scatter-add into slices 1 (users) and 4 (items)
  scatter_edges_kernel<<<scatterBlocks, 256, 0, stream>>>(
      uv_rows, uv_cols, uv_vals, s0, s3, s1, s4, N_EDGES);

  // 4) bias + LeakyReLU in place -> User_n (s1), Item_n (s4)
  bias_leaky_kernel<<<elemBlocks, 256, 0, stream>>>(s1, b1, s4, b2, n4);

  // 5) union GEMMs (K=256 concat) with fused bias + ReLU; overwrite temps
  gemm_wmma_f32<256, true><<<gemmBlocks, 128, 0, stream>>>(
      ufea, s1, Wu, bu, s0, N_NODES);
  gemm_wmma_f32<256, true><<<gemmBlocks, 128, 0, stream>>>(
      vfea, s4, Wi, bi, s3, N_NODES);

  // 6) passthrough copies
  copy2_kernel<<<elemBlocks, 256, 0, stream>>>(ufea, s2, vfea, s5, n4);
}